// TemporalAttention_50921132262135
// MI455X (gfx1250) — compile-verified
//
#include <hip/hip_runtime.h>
#include <hip/hip_bf16.h>

#define B_  16
#define C_  64
#define T_  32
#define HW_ 784
#define TT_ (T_ * T_)   // 1024

typedef float v2f __attribute__((ext_vector_type(2)));
typedef float v8f __attribute__((ext_vector_type(8)));

// ---------------------------------------------------------------------------
// Phase 1: per-(b,c) partial Gram, exploiting symmetry.
// 3 waves per WG: wave0 -> tile(0,0), wave1 -> tile(1,1), wave2 -> tile(1,0).
// Tile (0,1) is recovered by transpose in phase 2.  Diagonal-tile waves have
// identical A/B operand addresses (CSE -> 1 load per step).
// f32 A 16x4 layout: lane 0..15: M=lane, K={0,1}; lane 16..31: M=lane-16,
// K={2,3}.  f32 B 4x16 mirrors it with N = lane&15.  Both operands of
// slab @ slab^T are one aligned float2 load each.
// ---------------------------------------------------------------------------
__global__ __launch_bounds__(96) void gram_partial_kernel(
    const float* __restrict__ x, float* __restrict__ part) {
  const int bc   = blockIdx.x;                    // b*C + c
  const float* __restrict__ slab = x + (size_t)bc * (T_ * HW_);
  const int wave = threadIdx.x >> 5;              // 0,1,2
  const int lane = threadIdx.x & 31;
  const int rh   = lane & 15;                     // row within half-wave
  const int kh   = (lane >> 4) * 2;               // K sub-offset (0 or 2)
  const int mbase = wave ? 16 : 0;                // 0,16,16
  const int nbase = (wave == 1) ? 16 : 0;         // 0,16,0
  const float* __restrict__ ap = slab + (mbase + rh) * HW_ + kh;
  const float* __restrict__ bp = slab + (nbase + rh) * HW_ + kh;

  v8f acc = {};
#pragma unroll 4
  for (int k0 = 0; k0 < HW_; k0 += 4) {
    v2f a = *(const v2f*)(ap + k0);
    v2f b = *(const v2f*)(bp + k0);
    acc = __builtin_amdgcn_wmma_f32_16x16x4_f32(
        /*neg_a=*/false, a, /*neg_b=*/false, b,
        /*c_mod=*/(short)0, acc, /*reuse_a=*/false, /*reuse_b=*/false);
  }

  // C/D layout: VGPR r, lanes 0-15 -> M=r, lanes 16-31 -> M=8+r; N = lane&15.
  float* __restrict__ o = part + (size_t)bc * TT_;
  const int n    = nbase + rh;
  const int mrow = mbase + 8 * (lane >> 4);
#pragma unroll
  for (int r = 0; r < 8; ++r) o[(mrow + r) * T_ + n] = acc[r];
}

// ---------------------------------------------------------------------------
// Phase 2: reduce channel partials -> logits (reading the (0,1) tile as the
// transpose of the stored (1,0) tile), then softmax over the BATCH axis
// (per (q,k) across the 16 batches), faithful to the reference.
// ---------------------------------------------------------------------------
__global__ __launch_bounds__(256) void softmax_batch_kernel(
    const float* __restrict__ part, float* __restrict__ m) {
  const int qk = blockIdx.x * 256 + threadIdx.x;  // 0..1023
  if (qk >= TT_) return;
  const int q = qk >> 5;
  const int k = qk & 31;
  // Gram is symmetric per (b,c); tile q<16,k>=16 was not written -> transpose.
  const int idx = (q < 16 && k >= 16) ? (k * T_ + q) : qk;

  float logit[B_];
  float mx = -3.402823466e38f;
#pragma unroll
  for (int b = 0; b < B_; ++b) {
    float s = 0.f;
    for (int c = 0; c < C_; ++c)
      s += part[(size_t)(b * C_ + c) * TT_ + idx];
    logit[b] = s;
    mx = fmaxf(mx, s);
  }
  float den = 0.f;
#pragma unroll
  for (int b = 0; b < B_; ++b) {
    float e = __expf(logit[b] - mx);
    logit[b] = e;
    den += e;
  }
  const float inv = 1.f / den;
#pragma unroll
  for (int b = 0; b < B_; ++b) m[b * TT_ + qk] = logit[b] * inv;
}

// ---------------------------------------------------------------------------
// Phase 3: out_slab(32x784) = m_b(32x32) @ v_slab(32x784) per (b,c).
// Each wave owns N-tiles (nt = wave, wave+4, ...), preloads the A operands
// for BOTH M-tiles once (wave-invariant), and per K-step loads B once and
// feeds two WMMAs.  Inner loop: 2 x b32 load + 2 x wmma.
// ---------------------------------------------------------------------------
__global__ __launch_bounds__(128) void apply_kernel(
    const float* __restrict__ xv, const float* __restrict__ m,
    float* __restrict__ out) {
  const int bc = blockIdx.x;
  const int b  = bc >> 6;                          // / C_
  const float* __restrict__ vs = xv + (size_t)bc * (T_ * HW_);
  float* __restrict__ os       = out + (size_t)bc * (T_ * HW_);
  const float* __restrict__ mb = m + b * TT_;
  const int wave = threadIdx.x >> 5;
  const int lane = threadIdx.x & 31;
  const int rh   = lane & 15;
  const int kh   = (lane >> 4) * 2;

  // Preload A operands: a0 = rows 0..15 of m_b, a1 = rows 16..31, all K-steps.
  v2f a0[8], a1[8];
#pragma unroll
  for (int s = 0; s < 8; ++s) {
    a0[s] = *(const v2f*)(mb + (rh)      * T_ + 4 * s + kh);
    a1[s] = *(const v2f*)(mb + (16 + rh) * T_ + 4 * s + kh);
  }

  for (int nt = wave; nt < 49; nt += 4) {          // wave-uniform loop
    const int nbase = nt * 16;
    v8f acc0 = {}, acc1 = {};
#pragma unroll
    for (int s = 0; s < 8; ++s) {
      // B tile: B[k][n] = V[4s+k][nbase+n]; k = kh + {0,1}, n = lane&15.
      v2f bb;
      bb.x = vs[(4 * s + kh)     * HW_ + nbase + rh];
      bb.y = vs[(4 * s + kh + 1) * HW_ + nbase + rh];
      acc0 = __builtin_amdgcn_wmma_f32_16x16x4_f32(
          false, a0[s], false, bb, (short)0, acc0, false, false);
      acc1 = __builtin_amdgcn_wmma_f32_16x16x4_f32(
          false, a1[s], false, bb, (short)0, acc1, false, false);
    }
    const int mrow = 8 * (lane >> 4);
#pragma unroll
    for (int r = 0; r < 8; ++r) {
      os[(mrow + r)      * HW_ + nbase + rh] = acc0[r];
      os[(16 + mrow + r) * HW_ + nbase + rh] = acc1[r];
    }
  }
}

// ---------------------------------------------------------------------------
extern "C" void kernel_launch(void* const* d_in, const int* in_sizes, int n_in,
                              void* d_out, int out_size, void* d_ws, size_t ws_size,
                              hipStream_t stream) {
  const float* x  = (const float*)d_in[0];
  const float* xv = (const float*)d_in[1];
  float* out  = (float*)d_out;
  float* part = (float*)d_ws;                       // B*C*TT floats = 4 MB
  float* m    = part + (size_t)B_ * C_ * TT_;       // B*TT floats  = 64 KB

  gram_partial_kernel<<<B_ * C_, 96, 0, stream>>>(x, part);
  softmax_batch_kernel<<<(TT_ + 255) / 256, 256, 0, stream>>>(part, m);
  apply_kernel<<<B_ * C_, 128, 0, stream>>>(xv, m, out);
}